// MotionMatchingLoss_55396488184381
// MI455X (gfx1250) — compile-verified
//
#include <hip/hip_runtime.h>
#include <math.h>

typedef __attribute__((ext_vector_type(2))) float v2f;
typedef __attribute__((ext_vector_type(8))) float v8f;

#define MAXH 1080.0f
#define MAXW 1920.0f
#define BIGI 0x7f7fffff   // bits of FLT_MAX: acts as +inf sentinel for int-min

static __device__ __forceinline__ int imin2(int a, int b) { return a < b ? a : b; }
static __device__ __forceinline__ int imax2(int a, int b) { return a > b ? a : b; }

// Grid: 512 blocks (one per (n,t)), 256 threads (8 wave32).
// 512x512 pairwise d2 matrix via V_WMMA_F32_16X16X4_F32 with the K=4 embedding
//   d2[i][j] = [px,py,p2,1] . [-2qx,-2qy,1,q2]
// Wave w owns row-tiles 4w..4w+3; outer loop over col-tiles: one pipelined
// ds_load_b64 feeds 4 independent WMMAs.
// Min-reductions run in the INTEGER domain: clamp-to->=0 is v_max_i32(bits,0)
// (negative floats -> +0.0 bits), and for non-negative floats int order ==
// float order. This kills all v_max_num canonicalize traffic the FP minnum
// lowering was inserting.
__global__ __launch_bounds__(256)
void chamfer_wmma_kernel(const float* __restrict__ center,
                         const float* __restrict__ vel,
                         float* __restrict__ ws) {
    __shared__ v2f bvec[32 * 32];      // [cj][lane] B in per-lane WMMA layout
    __shared__ int colminH[8 * 1024];  // [wave][cj][lane] half-column mins (bits)
    __shared__ int rowpart[256 * 32];  // [ri*8+v][lane] row partial mins (bits)
    __shared__ float sred[256];

    const int b    = blockIdx.x;       // 0..511
    const int n    = b >> 5;
    const int t    = b & 31;
    const int tid  = threadIdx.x;
    const int wave = tid >> 5;
    const int lane = tid & 31;
    const int l16  = lane & 15;

    const float2* predC = (const float2*)(center + (size_t)((n * 33 + t)     * 512) * 2);
    const float2* predV = (const float2*)(vel    + (size_t)((n * 33 + t)     * 512) * 2);
    const float2* targC = (const float2*)(center + (size_t)((n * 33 + t + 1) * 512) * 2);

    // Stage B pre-swizzled: col j -> lanes 0-15 carry K=0,1 = (-2qx,-2qy),
    // lanes 16-31 carry K=2,3 = (1,q2).
    for (int j = tid; j < 512; j += 256) {
        float2 q = targC[j];
        int cj = j >> 4, l = j & 15;
        v2f lo; lo.x = -2.0f * q.x; lo.y = -2.0f * q.y;
        v2f hi; hi.x = 1.0f;        hi.y = q.x * q.x + q.y * q.y;
        bvec[cj * 32 + l]      = lo;
        bvec[cj * 32 + 16 + l] = hi;
    }
    __syncthreads();

    // Load the wave's 4 A tiles (A 16x4: lanes 0-15 K=0,1; lanes 16-31 K=2,3).
    v2f A[4];
#pragma unroll
    for (int rr = 0; rr < 4; ++rr) {
        int row = (wave * 4 + rr) * 16 + l16;
        float2 c = predC[row];
        float2 v = predV[row];
        float px = fminf(fmaxf(c.x + v.x, 0.0f), MAXH);
        float py = fminf(fmaxf(c.y + v.y, 0.0f), MAXW);
        float p2 = px * px + py * py;
        A[rr].x = (lane < 16) ? px : p2;
        A[rr].y = (lane < 16) ? py : 1.0f;
    }

    int rmin[4][8];
#pragma unroll
    for (int rr = 0; rr < 4; ++rr)
#pragma unroll
        for (int v = 0; v < 8; ++v) rmin[rr][v] = BIGI;

    // Hot loop: pipelined B load -> 4 WMMAs -> int clamp+min -> 1 strip store.
    v2f Bcur = bvec[lane];
    for (int cj = 0; cj < 32; ++cj) {
        int cnext = (cj < 31) ? cj + 1 : 31;
        v2f Bnext = bvec[cnext * 32 + lane];   // issue early; wait hides in mins

        int cmin = BIGI;
#pragma unroll
        for (int rr = 0; rr < 4; ++rr) {
            v8f c = {};
            c = __builtin_amdgcn_wmma_f32_16x16x4_f32(
                    false, A[rr], false, Bcur, (short)0, c, false, false);
#pragma unroll
            for (int v = 0; v < 8; ++v) {
                int cv = imax2(__float_as_int(c[v]), 0); // clamp d2 to >=0
                rmin[rr][v] = imin2(rmin[rr][v], cv);
                cmin        = imin2(cmin, cv);
            }
        }
        // lanes 0-15: min over rows M=0..7 of wave's tiles; lanes 16-31: M=8..15.
        colminH[wave * 1024 + cj * 32 + lane] = cmin;
        Bcur = Bnext;
    }

    // Dump row partial mins (once).
#pragma unroll
    for (int rr = 0; rr < 4; ++rr)
#pragma unroll
        for (int v = 0; v < 8; ++v)
            rowpart[((wave * 4 + rr) * 8 + v) * 32 + lane] = rmin[rr][v];
    __syncthreads();

    // Final phase: each thread resolves 2 rows + 2 cols (int mins, then sqrt).
    float acc = 0.0f;
#pragma unroll
    for (int s = 0; s < 2; ++s) {
        int r  = tid + s * 256;
        // Row r = ri*16 + m: half 0 (m<8) lives in lanes 0-15, half 1 in 16-31.
        int ri = r >> 4, m = r & 15;
        int base = ((ri * 8) + (m & 7)) * 32 + (m >> 3) * 16;
        int mn = rowpart[base];
#pragma unroll
        for (int u = 1; u < 16; ++u) mn = imin2(mn, rowpart[base + u]);
        acc += sqrtf(__int_as_float(mn));

        // Col r: min over 8 waves x 2 row-halves.
        int cj = r >> 4, l = r & 15;
        int cm = BIGI;
#pragma unroll
        for (int w = 0; w < 8; ++w) {
            cm = imin2(cm, colminH[w * 1024 + cj * 32 + l]);
            cm = imin2(cm, colminH[w * 1024 + cj * 32 + 16 + l]);
        }
        acc += sqrtf(__int_as_float(cm));
    }

    sred[tid] = acc;
    __syncthreads();
    for (int off = 128; off > 0; off >>= 1) {
        if (tid < off) sred[tid] += sred[tid + off];
        __syncthreads();
    }
    if (tid == 0) ws[b] = sred[0];   // fwd_sum + bwd_sum for this (n,t)
}

// Deterministic reduction of 512 per-batch sums -> scalar mean.
__global__ __launch_bounds__(256)
void chamfer_finalize_kernel(const float* __restrict__ ws, float* __restrict__ out) {
    __shared__ float s[256];
    const int tid = threadIdx.x;
    s[tid] = ws[tid] + ws[tid + 256];
    __syncthreads();
    for (int off = 128; off > 0; off >>= 1) {
        if (tid < off) s[tid] += s[tid + off];
        __syncthreads();
    }
    if (tid == 0) out[0] = s[0] * (1.0f / 524288.0f); // / (2*32*16*512)
}

extern "C" void kernel_launch(void* const* d_in, const int* in_sizes, int n_in,
                              void* d_out, int out_size, void* d_ws, size_t ws_size,
                              hipStream_t stream) {
    (void)in_sizes; (void)n_in; (void)out_size; (void)ws_size;
    const float* center = (const float*)d_in[0];
    const float* vel    = (const float*)d_in[1];
    float* ws  = (float*)d_ws;
    float* out = (float*)d_out;

    chamfer_wmma_kernel<<<512, 256, 0, stream>>>(center, vel, ws);
    chamfer_finalize_kernel<<<1, 256, 0, stream>>>(ws, out);
}